// SpikingFEAST_79912161509736
// MI455X (gfx1250) — compile-verified
//
#include <hip/hip_runtime.h>
#include <hip/hip_bf16.h>
#include <math.h>

// ---------------------------------------------------------------------------
// SpikingFEAST on MI455X (gfx1250, wave32)
//
// Roofline: 256MB read (weights) + 256MB write (new_weights) >> everything
// else  =>  memory bound, ~22us at 23.3 TB/s. Strategy: single streaming pass
// over weights that copies W -> new_weights (non-temporal, single-use stream)
// AND computes per-row squared distance to x via V_WMMA_F32_16X16X4_F32
// (diagonal of diff * diff^T), then a tiny O(N) epilogue kernel.
// ---------------------------------------------------------------------------

typedef __attribute__((ext_vector_type(2))) float v2f;
typedef __attribute__((ext_vector_type(4))) float v4f;
typedef __attribute__((ext_vector_type(8))) float v8f;

#define N_NEURONS 65536
#define DIM 1024
#define ROWS_PER_BLOCK 128   // 8 waves x 16 rows
#define KC 64                // K-chunk staged in LDS per iteration
#define LDS_PITCH 68         // floats; 16B-aligned rows, conflict-free b64 reads

#define F_OPEN     0.01f
#define LR_THRESH  0.01f
#define LR_WEIGHTS 0.01f
#define LR         1.0f

__global__ void feast_init_kernel(unsigned long long* __restrict__ argmin_key) {
    if (threadIdx.x == 0) *argmin_key = ~0ull;  // +inf key
}

__global__ __launch_bounds__(256)
void feast_dist_kernel(const float* __restrict__ x,
                       const float* __restrict__ weights,
                       float* __restrict__ out_weights,         // d_out + N
                       float* __restrict__ values,              // ws[0 .. N)
                       unsigned long long* __restrict__ argmin_key) {
    __shared__ float tile[ROWS_PER_BLOCK * LDS_PITCH];

    const int t    = threadIdx.x;
    const int lane = t & 31;
    const int wave = t >> 5;                   // 0..7
    const int row0 = blockIdx.x * ROWS_PER_BLOCK;

    // Loader mapping: 256 threads load a 16-row x 64-col slab per pass,
    // coalesced 16B along K.
    const int ldr_row = t >> 4;                // 0..15
    const int ldr_c4  = (t & 15) * 4;          // 0,4,...,60

    // WMMA consumer mapping: wave owns LDS rows [wave*16, wave*16+16).
    // Lane L supplies A[M = L%16][K = 2*(L>=16) + {0,1}]; for the symmetric
    // product, the B fragment is the identical register pair.
    const int myrow = wave * 16 + (lane & 15);
    const int koff  = (lane >> 4) * 2;         // 0 for lanes 0-15, 2 for 16-31
    const float* src = &tile[myrow * LDS_PITCH + koff];

    v8f acc = {};  // 16x16 f32 accumulator; diagonal = partial sum of squares

    for (int kk = 0; kk < DIM; kk += KC) {
        __syncthreads();  // protect LDS reuse from previous iteration
        #pragma unroll
        for (int pass = 0; pass < 8; ++pass) {
            const int r = pass * 16 + ldr_row;                 // 0..127
            const int g = (row0 + r) * DIM + kk + ldr_c4;
            // Single-use 512MB stream: non-temporal on both sides so it
            // doesn't thrash the 192MB L2.
            v4f w4 = __builtin_nontemporal_load((const v4f*)(weights + g));
            v4f x4 = *(const v4f*)(x + kk + ldr_c4);           // hot, keep RT
            __builtin_nontemporal_store(w4, (v4f*)(out_weights + g));
            *(v4f*)(&tile[r * LDS_PITCH + ldr_c4]) = w4 - x4;  // stage diff
        }
        __syncthreads();

        #pragma unroll
        for (int ks = 0; ks < KC / 4; ++ks) {
            v2f ab = *(const v2f*)(src + ks * 4);  // 8B-aligned ds_load_b64
            // D += diff_tile * diff_tile^T  (A and B fragments coincide)
            acc = __builtin_amdgcn_wmma_f32_16x16x4_f32(
                false, ab, false, ab, (short)0, acc, false, false);
        }
    }

    // Diagonal extraction per C/D layout: D[m][m] sits in VGPR (m&7) of lane m
    // (m<8) or lane m+16 (m>=8); owners here are lanes 0-7 and 16-23, so rows 8-15 read D[m][m-8].
    const bool owner = ((lane & 8) == 0);
    const int  di    = lane & 7;
    float ssq = acc[0];
    #pragma unroll
    for (int k = 1; k < 8; ++k) ssq = (di == k) ? acc[k] : ssq;  // cndmask chain
    if (owner) {
        const int rloc = di + ((lane >> 4) << 3);          // 0..15 within wave
        const int grow = row0 + wave * 16 + rloc;
        const float val = sqrtf(ssq);
        values[grow] = val;
        // Non-negative float: bit pattern is monotone -> pack for argmin,
        // low 32 bits = index gives first-index tie-break like jnp.argmin.
        unsigned long long key =
            ((unsigned long long)__float_as_uint(val) << 32) | (unsigned)grow;
        atomicMin(argmin_key, key);
    }
}

__global__ __launch_bounds__(256)
void feast_update_kernel(const float* __restrict__ x,
                         const float* __restrict__ reward_p,
                         const float* __restrict__ weights,
                         const float* __restrict__ thresholds,
                         const float* __restrict__ values,
                         const unsigned long long* __restrict__ argmin_key,
                         float* __restrict__ out_act,      // d_out
                         float* __restrict__ out_weights,  // d_out + N
                         float* __restrict__ out_thresh) { // d_out + N + N*D
    const float reward = *reward_p;
    const unsigned long long key = *argmin_key;
    const unsigned closest = (unsigned)(key & 0xffffffffu);
    const float vmin = __uint_as_float((unsigned)(key >> 32));

    const unsigned i = blockIdx.x * 256u + threadIdx.x;
    const float th = thresholds[i];
    const float v  = values[i];
    const bool eligible = (v <= th);
    const float th1 = eligible ? th : th + F_OPEN * reward;
    float nth = th1;
    if (i == closest) nth = th1 + (vmin - th1) * (LR_THRESH * LR * reward);
    out_thresh[i] = nth;
    out_act[i]    = (i == closest) ? 1.0f : 0.0f;

    // Winner weight-row rewrite: done by the block that owns `closest`
    // (overwrites the pass-through copy written by feast_dist_kernel).
    if (blockIdx.x == (closest >> 8)) {
        const size_t base = (size_t)closest * DIM;
        for (int j = threadIdx.x; j < DIM; j += 256) {
            const float wv = weights[base + j];
            out_weights[base + j] = wv + (x[j] - wv) * (LR_WEIGHTS * LR * reward);
        }
    }
}

extern "C" void kernel_launch(void* const* d_in, const int* in_sizes, int n_in,
                              void* d_out, int out_size, void* d_ws, size_t ws_size,
                              hipStream_t stream) {
    (void)in_sizes; (void)n_in; (void)out_size; (void)ws_size;

    const float* x          = (const float*)d_in[0];
    const float* reward     = (const float*)d_in[1];
    const float* weights    = (const float*)d_in[2];
    const float* thresholds = (const float*)d_in[3];

    float* out_act = (float*)d_out;
    float* out_w   = out_act + N_NEURONS;
    float* out_th  = out_w + (size_t)N_NEURONS * DIM;

    float* values = (float*)d_ws;                                   // N floats
    unsigned long long* argmin_key =
        (unsigned long long*)((char*)d_ws + (size_t)N_NEURONS * 4); // 8 bytes

    feast_init_kernel<<<1, 32, 0, stream>>>(argmin_key);
    feast_dist_kernel<<<N_NEURONS / ROWS_PER_BLOCK, 256, 0, stream>>>(
        x, weights, out_w, values, argmin_key);
    feast_update_kernel<<<N_NEURONS / 256, 256, 0, stream>>>(
        x, reward, weights, thresholds, values, argmin_key,
        out_act, out_w, out_th);
}